// RiemannianLIFNode_88811333746825
// MI455X (gfx1250) — compile-verified
//
#include <hip/hip_runtime.h>

// ---------------------------------------------------------------------------
// RiemannianLIFNode on MI455X (gfx1250).
// Memory-bound streaming problem (~560 MiB traffic, ~0.33 GFLOP => ~24us at
// 23.3 TB/s). No matmul structure -> no WMMA. We instead exploit the CDNA5
// async-to-LDS engine (ASYNCcnt) for deep, low-VGPR memory pipelining and
// non-temporal hints for the single-use streams.
// ---------------------------------------------------------------------------

typedef float f4 __attribute__((ext_vector_type(4)));
typedef float f2 __attribute__((ext_vector_type(2)));

#define TSTEPS     16
#define SCAN_BLOCK 256
// np.float32(np.exp(-0.05/2.0))
#define BETA 0.97530991202833267f

// Raw LDS byte offset of a generic pointer into __shared__ space
// (addrspace(3) pointers are 32-bit LDS offsets on amdgcn).
__device__ __forceinline__ unsigned lds_byte_offset(const void* p) {
  return (unsigned)(unsigned long long)(__attribute__((address_space(3))) const char*)p;
}

// ---------------------------------------------------------------------------
// Kernel 1: LIF scan.  Each thread owns one float4 column across all 16 time
// steps.  All 16 t-slices are launched as async global->LDS copies up front
// (ASYNCcnt = 16 in flight per wave), then drained in order with
// s_wait_asynccnt 15..0 while the recurrence runs in registers.
// ---------------------------------------------------------------------------
__global__ __launch_bounds__(SCAN_BLOCK) void lif_scan_async(
    const float* __restrict__ x, float* __restrict__ out, unsigned sliceF4) {
  __shared__ f4 stage_buf[TSTEPS * SCAN_BLOCK];

  const unsigned tid = threadIdx.x;
  const unsigned col = blockIdx.x * SCAN_BLOCK + tid;  // float4 index in slice
  if (col >= sliceF4) return;

  const f4* xg = (const f4*)x + col;
  f4*       og = (f4*)out + col;
  const unsigned lds0 = lds_byte_offset(&stage_buf[tid]);

  // Issue all 16 async slice loads back-to-back (tracked by ASYNCcnt,
  // completes in order).
#pragma unroll
  for (int t = 0; t < TSTEPS; ++t) {
    const f4*      gp = xg + (size_t)t * sliceF4;
    const unsigned la = lds0 + (unsigned)t * (SCAN_BLOCK * 16u);
    asm volatile("global_load_async_to_lds_b128 %0, %1, off"
                 :: "v"(la), "v"(gp)
                 : "memory");
  }

  const float omb = 1.0f - BETA;
  f4 m = {0.f, 0.f, 0.f, 0.f};

#define LIF_STAGE(T, CNT)                                                   \
  {                                                                         \
    asm volatile("s_wait_asynccnt " CNT ::: "memory");                      \
    f4 xv = stage_buf[(T) * SCAN_BLOCK + tid];                              \
    m = m * BETA + xv * omb;                                                \
    f4 sp;                                                                  \
    sp.x = m.x > 1.0f ? 1.0f : 0.0f;                                        \
    sp.y = m.y > 1.0f ? 1.0f : 0.0f;                                        \
    sp.z = m.z > 1.0f ? 1.0f : 0.0f;                                        \
    sp.w = m.w > 1.0f ? 1.0f : 0.0f;                                        \
    __builtin_nontemporal_store(sp, og + (size_t)(T) * sliceF4);            \
    m = m - sp;                                                             \
  }

  LIF_STAGE(0, "15")
  LIF_STAGE(1, "14")
  LIF_STAGE(2, "13")
  LIF_STAGE(3, "12")
  LIF_STAGE(4, "11")
  LIF_STAGE(5, "10")
  LIF_STAGE(6, "9")
  LIF_STAGE(7, "8")
  LIF_STAGE(8, "7")
  LIF_STAGE(9, "6")
  LIF_STAGE(10, "5")
  LIF_STAGE(11, "4")
  LIF_STAGE(12, "3")
  LIF_STAGE(13, "2")
  LIF_STAGE(14, "1")
  LIF_STAGE(15, "0")
#undef LIF_STAGE
}

// ---------------------------------------------------------------------------
// Kernel 2: Lorentz exponential map.  One wave32 per row of D=64, float2 per
// lane (coalesced 256B per row).  Minkowski sign (-u0*v0) folded into lane 0;
// wave-level reduction via shuffles (ds path on wave32).
// ---------------------------------------------------------------------------
__global__ __launch_bounds__(256) void lorentz_expmap(
    const float* __restrict__ v, const float* __restrict__ z,
    float* __restrict__ zo, unsigned nrows) {
  const unsigned lane = threadIdx.x & 31u;
  const unsigned row  = blockIdx.x * 8u + (threadIdx.x >> 5);
  if (row >= nrows) return;

  const size_t base = (size_t)row * 64u;
  f2 vv = __builtin_nontemporal_load((const f2*)(v + base) + lane);
  f2 zz = __builtin_nontemporal_load((const f2*)(z + base) + lane);

  const float sgn = (lane == 0u) ? -1.0f : 1.0f;  // <v,v>_L: -v0^2 + sum vi^2
  float local = sgn * vv.x * vv.x + vv.y * vv.y;

#pragma unroll
  for (int msk = 16; msk >= 1; msk >>= 1)
    local += __shfl_xor(local, msk, 32);

  const float vn = sqrtf(fmaxf(local, 1e-8f));
  const float ch = coshf(vn);
  const float sh = sinhf(vn) / vn;

  f2 o;
  o.x = ch * zz.x + sh * vv.x;
  o.y = ch * zz.y + sh * vv.y;
  __builtin_nontemporal_store(o, (f2*)(zo + base) + lane);
}

// ---------------------------------------------------------------------------
// Entry point.  Inputs (setup_inputs order): x_seq [T,N,D] f32,
// v_seq [N,D] f32, z_seq [N,D] f32.  Output: out [T,N,D] then z_output [N,D],
// concatenated flat in d_out.
// ---------------------------------------------------------------------------
extern "C" void kernel_launch(void* const* d_in, const int* in_sizes, int n_in,
                              void* d_out, int out_size, void* d_ws, size_t ws_size,
                              hipStream_t stream) {
  (void)n_in; (void)out_size; (void)d_ws; (void)ws_size;

  const float* x = (const float*)d_in[0];
  const float* v = (const float*)d_in[1];
  const float* z = (const float*)d_in[2];
  float* out = (float*)d_out;

  const unsigned nd      = (unsigned)in_sizes[1];  // N*D
  const unsigned sliceF4 = nd / 4u;                // float4 columns per slice
  const unsigned nrows   = nd / 64u;               // N (D = 64)

  float* z_out = out + (size_t)TSTEPS * nd;

  const dim3 b1(SCAN_BLOCK);
  const dim3 g1((sliceF4 + SCAN_BLOCK - 1) / SCAN_BLOCK);
  lif_scan_async<<<g1, b1, 0, stream>>>(x, out, sliceF4);

  const dim3 b2(256);
  const dim3 g2((nrows + 7u) / 8u);
  lorentz_expmap<<<g2, b2, 0, stream>>>(v, z, z_out, nrows);
}